// Network_p2_c2_41_21234318312193
// MI455X (gfx1250) — compile-verified
//
#include <hip/hip_runtime.h>
#include <cmath>

typedef float vf2 __attribute__((ext_vector_type(2)));
typedef float vf4 __attribute__((ext_vector_type(4)));

__device__ __forceinline__ float sigmoid_fast(float x) {
  // 1/(1+exp(-x)) via hw exp2 + hw rcp (v_exp_f32 / v_rcp_f32)
  float e = __builtin_amdgcn_exp2f(-1.44269504088896340736f * x);
  return __builtin_amdgcn_rcpf(1.0f + e);
}

__device__ __forceinline__ float sigmoid_acc(float x) {
  return 1.0f / (1.0f + expf(-x));
}

// s[i] = sigmoid(t[i]) elementwise (used for grid1, layout unchanged: float2 cells).
__global__ void sigmoid_table_kernel(const float* __restrict__ in,
                                     float* __restrict__ out, int n) {
  int stride = gridDim.x * blockDim.x;
  for (int i = blockIdx.x * blockDim.x + threadIdx.x; i < n; i += stride) {
    out[i] = sigmoid_acc(in[i]);
  }
}

// grid0: sigmoid + pad 3 channels -> 4 so each corner gather is one b128 load.
__global__ void sigmoid_pad3to4_kernel(const float* __restrict__ in,
                                       vf4* __restrict__ out, int ncells) {
  int stride = gridDim.x * blockDim.x;
  for (int i = blockIdx.x * blockDim.x + threadIdx.x; i < ncells; i += stride) {
    const float* p = in + 3u * (unsigned)i;
    vf4 o;
    o.x = sigmoid_acc(p[0]);
    o.y = sigmoid_acc(p[1]);
    o.z = sigmoid_acc(p[2]);
    o.w = 0.0f;
    out[i] = o;
  }
}

// Chained bilinear: key = bilerp2(grid1, x); rgb = bilerp3(grid0, key).
// PRE=true: tables pre-sigmoided in d_ws, grid0 padded to vf4 (b128 gathers).
// PRE=false: raw tables, inline hw sigmoid, 3-channel scalar gathers.
template <bool PRE>
__global__ void chained_bilinear_kernel(const float* __restrict__ x,
                                        const float* __restrict__ g1,
                                        const float* __restrict__ g0,
                                        float* __restrict__ out,
                                        int N, int R1, int R0) {
  const vf2* __restrict__ x2  = (const vf2*)x;
  const vf2* __restrict__ g1v = (const vf2*)g1;
  const vf4* __restrict__ g0v = (const vf4*)g0;   // valid only when PRE
  const float fR1 = (float)R1;
  const float fR0 = (float)R0;
  const int stride = gridDim.x * blockDim.x;

  for (int i = blockIdx.x * blockDim.x + threadIdx.x; i < N; i += stride) {
    // streaming read of coords ahead of time (global_prefetch_b8)
    if (i + stride < N) __builtin_prefetch(&x2[i + stride], 0, 1);

    vf2 uv = __builtin_nontemporal_load(&x2[i]);   // read-once stream: TH=NT

    // ---- level 1: (R1,R1,2) lookup -> uv key --------------------------------
    float su = uv.x * fR1;
    float sv = uv.y * fR1;
    float fu0 = floorf(su), fv0 = floorf(sv);
    float fu = su - fu0, fv = sv - fv0;
    int u0 = (int)fu0; if (u0 >= R1) u0 -= R1;   // wrap (inputs in [0,1))
    int v0 = (int)fv0; if (v0 >= R1) v0 -= R1;
    int u1 = u0 + 1;   if (u1 >= R1) u1 = 0;
    int v1 = v0 + 1;   if (v1 >= R1) v1 = 0;
    unsigned r0 = (unsigned)u0 * (unsigned)R1;
    unsigned r1 = (unsigned)u1 * (unsigned)R1;

    vf2 t00 = g1v[r0 + (unsigned)v0];
    vf2 t10 = g1v[r1 + (unsigned)v0];
    vf2 t01 = g1v[r0 + (unsigned)v1];
    vf2 t11 = g1v[r1 + (unsigned)v1];
    if (!PRE) {
      t00.x = sigmoid_fast(t00.x); t00.y = sigmoid_fast(t00.y);
      t10.x = sigmoid_fast(t10.x); t10.y = sigmoid_fast(t10.y);
      t01.x = sigmoid_fast(t01.x); t01.y = sigmoid_fast(t01.y);
      t11.x = sigmoid_fast(t11.x); t11.y = sigmoid_fast(t11.y);
    }
    float ofu = 1.0f - fu, ofv = 1.0f - fv;
    vf2 key = (t00 * ofu + t10 * fu) * ofv + (t01 * ofu + t11 * fu) * fv;

    // ---- level 0: (R0,R0,3) lookup -> rgb -----------------------------------
    su = key.x * fR0;
    sv = key.y * fR0;
    fu0 = floorf(su); fv0 = floorf(sv);
    fu = su - fu0; fv = sv - fv0;
    u0 = (int)fu0; if (u0 >= R0) u0 -= R0;
    v0 = (int)fv0; if (v0 >= R0) v0 -= R0;
    u1 = u0 + 1;   if (u1 >= R0) u1 = 0;
    v1 = v0 + 1;   if (v1 >= R0) v1 = 0;
    unsigned c00 = (unsigned)u0 * (unsigned)R0 + (unsigned)v0;
    unsigned c10 = (unsigned)u1 * (unsigned)R0 + (unsigned)v0;
    unsigned c01 = (unsigned)u0 * (unsigned)R0 + (unsigned)v1;
    unsigned c11 = (unsigned)u1 * (unsigned)R0 + (unsigned)v1;

    float r, g, b;
    ofu = 1.0f - fu; ofv = 1.0f - fv;
    if (PRE) {
      vf4 q00 = g0v[c00];                       // one b128 gather per corner
      vf4 q10 = g0v[c10];
      vf4 q01 = g0v[c01];
      vf4 q11 = g0v[c11];
      vf4 rgb = (q00 * ofu + q10 * fu) * ofv + (q01 * ofu + q11 * fu) * fv;
      r = rgb.x; g = rgb.y; b = rgb.z;
    } else {
      unsigned i00 = c00 * 3u, i10 = c10 * 3u, i01 = c01 * 3u, i11 = c11 * 3u;
      float c00r = sigmoid_fast(g0[i00]), c00g = sigmoid_fast(g0[i00 + 1]), c00b = sigmoid_fast(g0[i00 + 2]);
      float c10r = sigmoid_fast(g0[i10]), c10g = sigmoid_fast(g0[i10 + 1]), c10b = sigmoid_fast(g0[i10 + 2]);
      float c01r = sigmoid_fast(g0[i01]), c01g = sigmoid_fast(g0[i01 + 1]), c01b = sigmoid_fast(g0[i01 + 2]);
      float c11r = sigmoid_fast(g0[i11]), c11g = sigmoid_fast(g0[i11 + 1]), c11b = sigmoid_fast(g0[i11 + 2]);
      r = (c00r * ofu + c10r * fu) * ofv + (c01r * ofu + c11r * fu) * fv;
      g = (c00g * ofu + c10g * fu) * ofv + (c01g * ofu + c11g * fu) * fv;
      b = (c00b * ofu + c10b * fu) * ofv + (c01b * ofu + c11b * fu) * fv;
    }

    float* o = out + 3u * (unsigned)i;            // write-once stream: TH=NT
    __builtin_nontemporal_store(r, o);
    __builtin_nontemporal_store(g, o + 1);
    __builtin_nontemporal_store(b, o + 2);
  }
}

extern "C" void kernel_launch(void* const* d_in, const int* in_sizes, int n_in,
                              void* d_out, int out_size, void* d_ws, size_t ws_size,
                              hipStream_t stream) {
  const float* x  = (const float*)d_in[0];
  const float* g1 = (const float*)d_in[1];
  const float* g0 = (const float*)d_in[2];
  float* out = (float*)d_out;

  const int N  = in_sizes[0] / 2;
  const int n1 = in_sizes[1];                       // R1*R1*2
  const int n0 = in_sizes[2];                       // R0*R0*3
  const int R1 = (int)(sqrt((double)(n1 / 2)) + 0.5);
  const int R0 = (int)(sqrt((double)(n0 / 3)) + 0.5);
  const int ncells0 = R0 * R0;

  const int BLOCK = 256;                            // 8 wave32 per block
  int mainBlocks = (N + BLOCK - 1) / BLOCK;
  if (mainBlocks > 4096) mainBlocks = 4096;         // grid-stride, ~1M threads

  const size_t g0off = (((size_t)n1 * sizeof(float)) + 15) & ~(size_t)15;
  const size_t need  = g0off + (size_t)ncells0 * sizeof(vf4);
  if (ws_size >= need) {
    // Hoist sigmoid out of the hot loop and pad grid0 to 4 channels for b128 gathers.
    float* g1s = (float*)d_ws;
    vf4*   g0s = (vf4*)((char*)d_ws + g0off);
    int b1 = (n1 + BLOCK - 1) / BLOCK;      if (b1 > 4096) b1 = 4096;
    int b0 = (ncells0 + BLOCK - 1) / BLOCK; if (b0 > 4096) b0 = 4096;
    sigmoid_table_kernel<<<b1, BLOCK, 0, stream>>>(g1, g1s, n1);
    sigmoid_pad3to4_kernel<<<b0, BLOCK, 0, stream>>>(g0, g0s, ncells0);
    chained_bilinear_kernel<true><<<mainBlocks, BLOCK, 0, stream>>>(
        x, g1s, (const float*)g0s, out, N, R1, R0);
  } else {
    chained_bilinear_kernel<false><<<mainBlocks, BLOCK, 0, stream>>>(
        x, g1, g0, out, N, R1, R0);
  }
}